// MembershipDecoder_17798344475039
// MI455X (gfx1250) — compile-verified
//
#include <hip/hip_runtime.h>
#include <hip/hip_bf16.h>
#include <math.h>

// MembershipDecoder for MI455X (gfx1250), wave32, WMMA f32 16x16x4.
//
// Factorized algorithm (exact algebra, fp32 throughout):
//   seq_dec = relu(seq_feat@Ws+bs); col_dec = relu(col_feat@Wc+bc)
//   A = seq_dec@Wm ; B = col_dec@Wm + bm
//   a' = A - mean(A,h); b' = B - mean(B,h)
//   var[i,j] = varA[i] + varB[j] + (2/H) * <a'_i, b'_j>        (WMMA cross GEMM)
//   raw[i,j] = (p[i]+q[j]) * rsqrt(var+eps) + c
//   E = exp(raw); out = Mc + Ms - Mc*Ms (row softmax + segment-ragged norm).
//
// v3 changes vs v2 (driven by disasm): the K-loop showed `s_wait_loadcnt 0x0`
// immediately before almost every v_wmma (no in-wave latency hiding). The
// fragment loads are now explicitly software-pipelined: iteration k issues
// the loads for k+4 BEFORE its WMMAs, with the last step peeled, so each wait
// covers loads issued a full iteration earlier.

typedef __attribute__((ext_vector_type(2))) float v2f;
typedef __attribute__((ext_vector_type(8))) float v8f;

#define LN_EPS 1e-3f

// ---------------------------------------------------------------------------
// NT 16x16 fp32 WMMA tiles per wave (shared A fragment, double-buffered).
// A:  MxK row-major (16 rows used).   BT: NxK row-major (B transposed).
// Layouts per CDNA5 ISA 7.12.2:
//   A 16x4 f32 : lanes 0-15 -> M=lane, K={0,1}; lanes 16-31 -> M=lane-16, K={2,3}
//   B 4x16 f32 : lanes 0-15 -> N=lane, K={0,1}; lanes 16-31 -> N=lane-16, K={2,3}
//   C/D 16x16  : VGPR v -> M=v (lanes 0-15) / M=v+8 (lanes 16-31), N=lane&15
// ---------------------------------------------------------------------------
template <int NT>
__device__ __forceinline__ void wmma_strip(const float* __restrict__ A, int lda,
                                           const float* __restrict__ BT, int ldb,
                                           int K, v8f acc[NT]) {
  const int lane = threadIdx.x & 31;
  const int r    = lane & 15;
  const int kk   = (lane >> 4) << 1;   // 0 or 2
  const float* __restrict__ ap = A + r * lda + kk;
  const float* __restrict__ bp = BT + r * ldb + kk;

  // prologue: fragments for k = 0
  v2f a = *(const v2f*)ap;
  v2f b[NT];
#pragma unroll
  for (int t = 0; t < NT; ++t) b[t] = *(const v2f*)(bp + t * 16 * ldb);

  int k = 0;
  for (; k + 4 < K; k += 4) {
    // prefetch fragments for k+4 before consuming this step's
    const v2f an = *(const v2f*)(ap + k + 4);
    v2f bn[NT];
#pragma unroll
    for (int t = 0; t < NT; ++t) bn[t] = *(const v2f*)(bp + t * 16 * ldb + k + 4);
    // 8 args: (neg_a, A, neg_b, B, c_mod, C, reuse_a, reuse_b)
#pragma unroll
    for (int t = 0; t < NT; ++t)
      acc[t] = __builtin_amdgcn_wmma_f32_16x16x4_f32(false, a, false, b[t], (short)0,
                                                     acc[t], false, false);
    a = an;
#pragma unroll
    for (int t = 0; t < NT; ++t) b[t] = bn[t];
  }
  // epilogue: last K-step
#pragma unroll
  for (int t = 0; t < NT; ++t)
    acc[t] = __builtin_amdgcn_wmma_f32_16x16x4_f32(false, a, false, b[t], (short)0,
                                                   acc[t], false, false);
}

// out[c*R + r] = in[r*C + c]   (pre-transpose of weight matrices)
__global__ void k_transpose(const float* __restrict__ in, float* __restrict__ out,
                            int R, int C) {
  const int idx = blockIdx.x * blockDim.x + threadIdx.x;
  if (idx >= R * C) return;
  const int r = idx / C;
  const int c = idx - r * C;
  out[c * R + r] = in[r * C + c];
}

// D[16 x 64 strip] = op(A[16xK] @ B[Kx64] + bias); B given transposed (64 x K).
template <bool RELU>
__global__ void k_gemm_bias(const float* __restrict__ A, const float* __restrict__ BT,
                            const float* __restrict__ bias, float* __restrict__ D,
                            int lda, int ldb, int ldd, int K) {
  const int i0 = blockIdx.x * 16;
  const int j0 = blockIdx.y * 64;
  v8f acc[4] = {};
  wmma_strip<4>(A + i0 * lda, lda, BT + j0 * ldb, ldb, K, acc);
  const int lane  = threadIdx.x & 31;
  const int rbase = i0 + (lane >> 4) * 8;
#pragma unroll
  for (int t = 0; t < 4; ++t) {
    const int col  = j0 + t * 16 + (lane & 15);
    const float bv = bias ? bias[col] : 0.0f;
#pragma unroll
    for (int v = 0; v < 8; ++v) {
      float x = acc[t][v] + bv;
      if (RELU) x = fmaxf(x, 0.0f);
      D[(rbase + v) * ldd + col] = x;
    }
  }
}

// Per-row LayerNorm prep over H=128: center M in place, emit var and
// p = sum_h gamma*Wo*centered. Block i handles row i with 128 threads.
// If cconst != null, block 0 also computes c = <beta,Wo> + bo.
__global__ void k_row_stats(float* __restrict__ M, const float* __restrict__ addvec,
                            const float* __restrict__ gamma, const float* __restrict__ Wo,
                            float* __restrict__ varOut, float* __restrict__ pOut,
                            const float* __restrict__ beta, const float* __restrict__ bo,
                            float* __restrict__ cconst) {
  __shared__ float red[128];
  const int i = blockIdx.x;
  const int h = threadIdx.x;
  float x = M[i * 128 + h] + (addvec ? addvec[h] : 0.0f);

  red[h] = x;
  __syncthreads();
  for (int s = 64; s > 0; s >>= 1) { if (h < s) red[h] += red[h + s]; __syncthreads(); }
  const float mu = red[0] * (1.0f / 128.0f);
  __syncthreads();

  const float c = x - mu;
  M[i * 128 + h] = c;

  red[h] = c * c;
  __syncthreads();
  for (int s = 64; s > 0; s >>= 1) { if (h < s) red[h] += red[h + s]; __syncthreads(); }
  const float var = red[0] * (1.0f / 128.0f);
  __syncthreads();

  red[h] = c * gamma[h] * Wo[h];
  __syncthreads();
  for (int s = 64; s > 0; s >>= 1) { if (h < s) red[h] += red[h + s]; __syncthreads(); }
  if (h == 0) { varOut[i] = var; pOut[i] = red[0]; }

  if (cconst && i == 0) {
    __syncthreads();
    red[h] = beta[h] * Wo[h];
    __syncthreads();
    for (int s = 64; s > 0; s >>= 1) { if (h < s) red[h] += red[h + s]; __syncthreads(); }
    if (h == 0) cconst[0] = red[0] + bo[0];
  }
}

// Cross term + fused LN/Dense/exp epilogue: 16x64 (i,j) strip per wave.
__global__ void k_cross(const float* __restrict__ aC, const float* __restrict__ bC,
                        const float* __restrict__ varA, const float* __restrict__ pA,
                        const float* __restrict__ varB, const float* __restrict__ qB,
                        const float* __restrict__ cconst, float* __restrict__ E,
                        int H, int NCOL) {
  const int i0 = blockIdx.x * 16;
  const int j0 = blockIdx.y * 64;
  v8f acc[4] = {};
  wmma_strip<4>(aC + i0 * H, H, bC + j0 * H, H, H, acc);  // acc = <a'_i, b'_j>
  const int lane  = threadIdx.x & 31;
  const int rbase = i0 + (lane >> 4) * 8;
  const float cc = cconst[0];
  const float two_over_H = 2.0f / (float)H;
#pragma unroll
  for (int t = 0; t < 4; ++t) {
    const int col  = j0 + t * 16 + (lane & 15);
    const float vb = varB[col];
    const float qv = qB[col];
#pragma unroll
    for (int v = 0; v < 8; ++v) {
      const int row = rbase + v;
      const float var = varA[row] + vb + acc[t][v] * two_over_H;
      const float raw = (pA[row] + qv) * rsqrtf(var + LN_EPS) + cc;
      E[row * NCOL + col] = expf(raw);
    }
  }
}

// rowsum[i] = sum_j E[i,j]  (block per row, deterministic tree reduce)
__global__ void k_rowsum(const float* __restrict__ E, float* __restrict__ rowsum, int NCOL) {
  __shared__ float red[128];
  const int i = blockIdx.x;
  const int t = threadIdx.x;
  float s = 0.0f;
  for (int j = t; j < NCOL; j += 128) s += E[i * NCOL + j];
  red[t] = s;
  __syncthreads();
  for (int st = 64; st > 0; st >>= 1) { if (t < st) red[t] += red[t + st]; __syncthreads(); }
  if (t == 0) rowsum[i] = red[0];
}

// segsum[s][j] = sum_{i in segment s} E[i,j]; serial over rows -> deterministic.
__global__ void k_segsum(const float* __restrict__ E, const int* __restrict__ lens,
                         float* __restrict__ segsum, int NCOL) {
  const int s = blockIdx.x;
  const int j = blockIdx.y * 128 + threadIdx.x;
  if (j >= NCOL) return;
  int off = 0;
  for (int t = 0; t < s; ++t) off += lens[t];
  const int len = lens[s];
  float acc = 0.0f;
  for (int i = 0; i < len; ++i) acc += E[(off + i) * NCOL + j];
  segsum[s * NCOL + j] = acc;
}

// out = Mc + Ms - Mc*Ms
__global__ void k_final(const float* __restrict__ E, const float* __restrict__ rowsum,
                        const float* __restrict__ segsum, const int* __restrict__ lens,
                        int n_seq, int NCOL, float* __restrict__ out, int total) {
  const int idx = blockIdx.x * blockDim.x + threadIdx.x;
  if (idx >= total) return;
  const int i = idx / NCOL;
  const int j = idx - i * NCOL;
  const float e  = E[idx];
  const float mc = e / rowsum[i];
  int off = 0, s = 0;
  for (int t = 0; t < n_seq; ++t) {
    const int l = lens[t];
    if (i >= off + l) { off += l; ++s; }
  }
  const float ms = e / segsum[s * NCOL + j];
  out[idx] = mc + ms - mc * ms;
}

extern "C" void kernel_launch(void* const* d_in, const int* in_sizes, int n_in,
                              void* d_out, int out_size, void* d_ws, size_t ws_size,
                              hipStream_t stream) {
  const float* seq_feat = (const float*)d_in[0];
  const float* col_feat = (const float*)d_in[1];
  const int*   lens     = (const int*)d_in[2];
  const float* Ws       = (const float*)d_in[3];
  const float* bs       = (const float*)d_in[4];
  const float* Wc       = (const float*)d_in[5];
  const float* bc       = (const float*)d_in[6];
  const float* Wm       = (const float*)d_in[7];
  const float* bm       = (const float*)d_in[8];
  const float* gamma    = (const float*)d_in[9];
  const float* beta     = (const float*)d_in[10];
  const float* Wo       = (const float*)d_in[11];
  const float* bo       = (const float*)d_in[12];
  float* out            = (float*)d_out;

  const int H    = in_sizes[4];            // 128
  const int D    = in_sizes[3] / H;        // 128
  const int NPOS = in_sizes[0] / D;        // 1024
  const int NCOL = in_sizes[1] / D;        // 512
  const int NSEQ = in_sizes[2];            // 8

  // Workspace layout (fp32), ~3.7 MB total.
  float* ws     = (float*)d_ws;
  float* tmp    = ws;                      // NPOS*H : relu'd decodes (reused)
  float* aC     = tmp + NPOS * H;          // NPOS*H : A, then centered a'
  float* bC     = aC + NPOS * H;           // NCOL*H : B, then centered b'
  float* E      = bC + NCOL * H;           // NPOS*NCOL : exp(raw)
  float* varA   = E + NPOS * NCOL;
  float* pA     = varA + NPOS;
  float* varB   = pA + NPOS;
  float* qB     = varB + NCOL;
  float* rowsum = qB + NCOL;
  float* segsum = rowsum + NPOS;
  float* cconst = segsum + NSEQ * NCOL;
  float* WsT    = cconst + 1;              // H*D
  float* WcT    = WsT + H * D;             // H*D
  float* WmT    = WcT + H * D;             // H*H

  // 0) pre-transpose weights so all WMMA fragment loads are contiguous v2f
  k_transpose<<<(D * H + 255) / 256, 256, 0, stream>>>(Ws, WsT, D, H);
  k_transpose<<<(D * H + 255) / 256, 256, 0, stream>>>(Wc, WcT, D, H);
  k_transpose<<<(H * H + 255) / 256, 256, 0, stream>>>(Wm, WmT, H, H);

  // 1) seq_dec = relu(seq_feat@Ws+bs); A = seq_dec@Wm
  k_gemm_bias<true><<<dim3(NPOS / 16, H / 64), 32, 0, stream>>>(seq_feat, WsT, bs, tmp, D, D, H, D);
  k_gemm_bias<false><<<dim3(NPOS / 16, H / 64), 32, 0, stream>>>(tmp, WmT, nullptr, aC, H, H, H, H);
  // 2) col_dec = relu(col_feat@Wc+bc); B = col_dec@Wm (bm folded into col stats)
  k_gemm_bias<true><<<dim3(NCOL / 16, H / 64), 32, 0, stream>>>(col_feat, WcT, bc, tmp, D, D, H, D);
  k_gemm_bias<false><<<dim3(NCOL / 16, H / 64), 32, 0, stream>>>(tmp, WmT, nullptr, bC, H, H, H, H);
  // 3) per-row LN statistics (centers aC/bC in place)
  k_row_stats<<<NPOS, 128, 0, stream>>>(aC, nullptr, gamma, Wo, varA, pA, beta, bo, cconst);
  k_row_stats<<<NCOL, 128, 0, stream>>>(bC, bm, gamma, Wo, varB, qB, nullptr, nullptr, nullptr);
  // 4) cross GEMM + fused epilogue -> E
  k_cross<<<dim3(NPOS / 16, NCOL / 64), 32, 0, stream>>>(aC, bC, varA, pA, varB, qB, cconst, E, H, NCOL);
  // 5) normalizers
  k_rowsum<<<NPOS, 128, 0, stream>>>(E, rowsum, NCOL);
  k_segsum<<<dim3(NSEQ, (NCOL + 127) / 128), 128, 0, stream>>>(E, lens, segsum, NCOL);
  // 6) combine
  const int total = NPOS * NCOL;
  k_final<<<(total + 255) / 256, 256, 0, stream>>>(E, rowsum, segsum, lens, NSEQ, NCOL, out, total);
}